// Label_Fusion_Layer_for_Token_40553081208895
// MI455X (gfx1250) — compile-verified
//
#include <hip/hip_runtime.h>
#include <math.h>

// ---------------------------------------------------------------------------
// Label-fusion layer for MI455X (gfx1250), wave32 + V_WMMA_F32_16X16X4_F32.
// B=8 S=256 H=768 L=20 LS=8 E=300.  Dominant cost: fused[40960x3072] @ W5^T.
// ---------------------------------------------------------------------------

typedef __attribute__((ext_vector_type(2))) float v2f;
typedef __attribute__((ext_vector_type(4))) float v4f;
typedef __attribute__((ext_vector_type(8))) float v8f;

#define NEG_BIG (-10000.0f)

#define TM 128   // block tile M
#define TN 64    // block tile N
#define TK 32    // k chunk
#define TKP 34   // LDS row stride (pad 2 floats -> spread 64 banks, keeps 8B align)

// D = A x B + C layout (16x16x4 f32, wave32):
//   A frag (v2f): lane l, reg r -> A[m = l&15][k = 2*(l>>4) + r]
//   B frag (v2f): lane l, reg r -> B[k = 2*(l>>4) + r][n = l&15]   (B[k][n] = W[n][k])
//   C/D  (v8f):  lane l, reg r -> C[m = r + 8*(l>>4)][n = l&15]

// ---------------------------------------------------------------------------
// Generic  C[M,N] = A[M,K] @ W[N,K]^T   (torch Linear), all f32 row-major.
// Requires K % 4 == 0 (true for 768, 300).  M/N/K-tail guarded via zero pad.
// ---------------------------------------------------------------------------
__global__ __launch_bounds__(256) void gemm_nt_f32_wmma(
    const float* __restrict__ A, const float* __restrict__ W,
    float* __restrict__ C, int M, int N, int K)
{
    __shared__ float As[TM][TKP];
    __shared__ float Ws[TN][TKP];

    const int tid  = threadIdx.x;
    const int lane = tid & 31, wid = tid >> 5;
    const int wm   = wid >> 1, wn  = wid & 1;      // 4x2 wave grid
    const int ll   = lane & 15, lh = lane >> 4;
    const int m0   = blockIdx.x * TM, n0 = blockIdx.y * TN;

    const v8f vzero = {0.f,0.f,0.f,0.f,0.f,0.f,0.f,0.f};
    v8f acc[2][2];
    acc[0][0]=vzero; acc[0][1]=vzero; acc[1][0]=vzero; acc[1][1]=vzero;

    for (int k0 = 0; k0 < K; k0 += TK) {
        __syncthreads();
        // stage A tile 128x32 (4 passes, one float4 per thread per pass)
        #pragma unroll
        for (int p = 0; p < 4; ++p) {
            int r = p*32 + (tid >> 3);
            int c = (tid & 7) << 2;
            int gm = m0 + r, gk = k0 + c;
            v4f v = {0.f,0.f,0.f,0.f};
            if (gm < M && gk < K) v = *(const v4f*)(A + (size_t)gm*K + gk);
            *(v4f*)&As[r][c] = v;
        }
        // stage W tile 64x32 (2 passes)
        #pragma unroll
        for (int p = 0; p < 2; ++p) {
            int r = p*32 + (tid >> 3);
            int c = (tid & 7) << 2;
            int gn = n0 + r, gk = k0 + c;
            v4f v = {0.f,0.f,0.f,0.f};
            if (gn < N && gk < K) v = *(const v4f*)(W + (size_t)gn*K + gk);
            *(v4f*)&Ws[r][c] = v;
        }
        __syncthreads();

        #pragma unroll
        for (int kk = 0; kk < 8; ++kk) {
            const int kc = 4*kk + 2*lh;
            v2f a0 = *(const v2f*)&As[wm*32      + ll][kc];
            v2f a1 = *(const v2f*)&As[wm*32 + 16 + ll][kc];
            v2f b0 = *(const v2f*)&Ws[wn*32      + ll][kc];
            v2f b1 = *(const v2f*)&Ws[wn*32 + 16 + ll][kc];
            acc[0][0] = __builtin_amdgcn_wmma_f32_16x16x4_f32(false,a0,false,b0,(short)0,acc[0][0],false,false);
            acc[0][1] = __builtin_amdgcn_wmma_f32_16x16x4_f32(false,a0,false,b1,(short)0,acc[0][1],false,false);
            acc[1][0] = __builtin_amdgcn_wmma_f32_16x16x4_f32(false,a1,false,b0,(short)0,acc[1][0],false,false);
            acc[1][1] = __builtin_amdgcn_wmma_f32_16x16x4_f32(false,a1,false,b1,(short)0,acc[1][1],false,false);
        }
    }

    #pragma unroll
    for (int mt = 0; mt < 2; ++mt)
        #pragma unroll
        for (int nt = 0; nt < 2; ++nt) {
            int gn = n0 + wn*32 + nt*16 + ll;
            #pragma unroll
            for (int r = 0; r < 8; ++r) {
                int gm = m0 + wm*32 + mt*16 + lh*8 + r;
                if (gm < M && gn < N) C[(size_t)gm*N + gn] = acc[mt][nt][r];
            }
        }
}

// ---------------------------------------------------------------------------
// Per-(b,s,l): masked softmax over LS=8 -> attn; b_in = max + (1-imask)*NEG.
// scores flat: row = bs*20 + l, 8 contiguous entries.
// ---------------------------------------------------------------------------
__global__ __launch_bounds__(256) void attn_softmax_kernel(
    const float* __restrict__ scores, const float* __restrict__ lmask,
    const float* __restrict__ imask, float* __restrict__ attn,
    float* __restrict__ b_in)
{
    int row = blockIdx.x * 256 + threadIdx.x;
    if (row >= 8*256*20) return;
    int bs = row / 20;
    int l  = row - bs*20;

    float v[8];
    float mx = -3.4e38f;
    #pragma unroll
    for (int t = 0; t < 8; ++t) {
        float s = scores[(size_t)row*8 + t] + (1.0f - lmask[l*8 + t]) * NEG_BIG;
        v[t] = s;
        mx = fmaxf(mx, s);
    }
    float sum = 0.f;
    #pragma unroll
    for (int t = 0; t < 8; ++t) { float e = __expf(v[t] - mx); v[t] = e; sum += e; }
    float inv = 1.0f / sum;
    #pragma unroll
    for (int t = 0; t < 8; ++t) attn[(size_t)row*8 + t] = v[t] * inv;

    b_in[row] = mx + (1.0f - imask[bs]) * NEG_BIG;
}

// ---------------------------------------------------------------------------
// Per-(b,l): softmax over S=256 of b_in, then q2c[b,l,:] = sum_s w[s]*tkn[b,s,:]
// One workgroup (256 threads = one thread per s) per (b,l).
// ---------------------------------------------------------------------------
__global__ __launch_bounds__(256) void q2c_kernel(
    const float* __restrict__ b_in, const float* __restrict__ tkn,
    float* __restrict__ q2c)
{
    __shared__ float red[256];
    __shared__ float wgt[256];
    const int tid = threadIdx.x;
    const int b = blockIdx.x / 20;
    const int l = blockIdx.x - b*20;

    float x = b_in[(size_t)(b*256 + tid)*20 + l];
    red[tid] = x; __syncthreads();
    for (int st = 128; st > 0; st >>= 1) {
        if (tid < st) red[tid] = fmaxf(red[tid], red[tid + st]);
        __syncthreads();
    }
    float mx = red[0]; __syncthreads();
    float e = __expf(x - mx);
    red[tid] = e; __syncthreads();
    for (int st = 128; st > 0; st >>= 1) {
        if (tid < st) red[tid] += red[tid + st];
        __syncthreads();
    }
    float sm = red[0]; __syncthreads();
    wgt[tid] = e / sm; __syncthreads();

    for (int h = tid; h < 768; h += 256) {
        float acc = 0.f;
        const float* tp = tkn + (size_t)b*256*768 + h;
        #pragma unroll 4
        for (int s = 0; s < 256; ++s) acc += wgt[s] * tp[(size_t)s*768];
        q2c[(size_t)(b*20 + l)*768 + h] = acc;
    }
}

// ---------------------------------------------------------------------------
// Main GEMM:  out[m, o] = tanh( fused[m, :3072] @ W5[o, :3072] + b5[o] )
//   m = bs*20 + l  (M = 40960), N = 768, K = 3072.
// fused is generated on the fly into LDS per 32-wide K chunk:
//   sec0: tkn[bs,j]   sec1: c2q   sec2: tkn*c2q   sec3: tkn*q2c[b,l,j]
//   with c2q[m, j] = sum_{t<8} attn[m,t] * lab[l,t,j]   (attn cached in LDS).
// M,N,K all divide tile sizes exactly -> no guards on the hot path.
// ---------------------------------------------------------------------------
__global__ __launch_bounds__(256) void fused_w5_tanh_wmma(
    const float* __restrict__ tkn,  const float* __restrict__ lab,
    const float* __restrict__ attn, const float* __restrict__ q2c,
    const float* __restrict__ W5,   const float* __restrict__ b5,
    float* __restrict__ out)
{
    __shared__ float As[TM][TKP];
    __shared__ float Ws[TN][TKP];
    __shared__ float aBlk[TM][8];

    const int tid  = threadIdx.x;
    const int lane = tid & 31, wid = tid >> 5;
    const int wm   = wid >> 1, wn  = wid & 1;
    const int ll   = lane & 15, lh = lane >> 4;
    const int m0   = blockIdx.x * TM, n0 = blockIdx.y * TN;

    // cache this block's 128x8 attention weights once
    {
        int r = tid >> 1, c = (tid & 1) << 2;
        *(v4f*)&aBlk[r][c] = *(const v4f*)(attn + (size_t)(m0 + r)*8 + c);
    }

    const v8f vzero = {0.f,0.f,0.f,0.f,0.f,0.f,0.f,0.f};
    v8f acc[2][2];
    acc[0][0]=vzero; acc[0][1]=vzero; acc[1][0]=vzero; acc[1][1]=vzero;

    for (int k0 = 0; k0 < 3072; k0 += TK) {
        const int sec = k0 / 768;
        const int j0  = k0 - sec*768;
        __syncthreads();
        // stage W5 tile 64x32 (W5 is [768][3072] row-major == [n][k])
        #pragma unroll
        for (int p = 0; p < 2; ++p) {
            int r = p*32 + (tid >> 3);
            int c = (tid & 7) << 2;
            *(v4f*)&Ws[r][c] = *(const v4f*)(W5 + (size_t)(n0 + r)*3072 + (k0 + c));
        }
        // generate fused A tile 128x32
        #pragma unroll
        for (int p = 0; p < 4; ++p) {
            int r = p*32 + (tid >> 3);
            int c = (tid & 7) << 2;
            int m  = m0 + r;
            int bs = m / 20;
            int l  = m - bs*20;
            int j  = j0 + c;
            v4f v;
            if (sec == 0) {
                v = *(const v4f*)(tkn + (size_t)bs*768 + j);
            } else if (sec <= 2) {
                v4f cq = {0.f,0.f,0.f,0.f};
                const float* labRow = lab + (size_t)l*8*768 + j;
                #pragma unroll
                for (int t = 0; t < 8; ++t) {
                    float w  = aBlk[r][t];
                    v4f  lv  = *(const v4f*)(labRow + (size_t)t*768);
                    cq += w * lv;
                }
                if (sec == 1) v = cq;
                else {
                    v4f t4 = *(const v4f*)(tkn + (size_t)bs*768 + j);
                    v = t4 * cq;
                }
            } else {
                int b  = bs >> 8;                       // bs / S
                v4f t4 = *(const v4f*)(tkn + (size_t)bs*768 + j);
                v4f q4 = *(const v4f*)(q2c + (size_t)(b*20 + l)*768 + j);
                v = t4 * q4;
            }
            *(v4f*)&As[r][c] = v;
        }
        __syncthreads();

        #pragma unroll
        for (int kk = 0; kk < 8; ++kk) {
            const int kc = 4*kk + 2*lh;
            v2f a0 = *(const v2f*)&As[wm*32      + ll][kc];
            v2f a1 = *(const v2f*)&As[wm*32 + 16 + ll][kc];
            v2f b0 = *(const v2f*)&Ws[wn*32      + ll][kc];
            v2f b1 = *(const v2f*)&Ws[wn*32 + 16 + ll][kc];
            acc[0][0] = __builtin_amdgcn_wmma_f32_16x16x4_f32(false,a0,false,b0,(short)0,acc[0][0],false,false);
            acc[0][1] = __builtin_amdgcn_wmma_f32_16x16x4_f32(false,a0,false,b1,(short)0,acc[0][1],false,false);
            acc[1][0] = __builtin_amdgcn_wmma_f32_16x16x4_f32(false,a1,false,b0,(short)0,acc[1][0],false,false);
            acc[1][1] = __builtin_amdgcn_wmma_f32_16x16x4_f32(false,a1,false,b1,(short)0,acc[1][1],false,false);
        }
    }

    #pragma unroll
    for (int mt = 0; mt < 2; ++mt)
        #pragma unroll
        for (int nt = 0; nt < 2; ++nt) {
            int gn = n0 + wn*32 + nt*16 + ll;
            float bias = b5[gn];
            #pragma unroll
            for (int r = 0; r < 8; ++r) {
                int gm = m0 + wm*32 + mt*16 + lh*8 + r;
                out[(size_t)gm*768 + gn] = tanhf(acc[mt][nt][r] + bias);
            }
        }
}

// ---------------------------------------------------------------------------
extern "C" void kernel_launch(void* const* d_in, const int* in_sizes, int n_in,
                              void* d_out, int out_size, void* d_ws, size_t ws_size,
                              hipStream_t stream)
{
    (void)in_sizes; (void)n_in; (void)out_size; (void)ws_size;

    const float* token = (const float*)d_in[0];   // [8,256,768]
    const float* label = (const float*)d_in[1];   // [20,8,300]
    const float* imask = (const float*)d_in[2];   // [8,256]
    const float* lmask = (const float*)d_in[3];   // [20,8]
    const float* W1    = (const float*)d_in[4];   // [768,768]
    const float* W2    = (const float*)d_in[5];   // [768,300]
    const float* W5    = (const float*)d_in[6];   // [768,3072]
    const float* b5    = (const float*)d_in[7];   // [768]
    float* out = (float*)d_out;                   // [8,256,20,768]

    // workspace layout (floats): ~9.6 MB total
    float* ws     = (float*)d_ws;
    float* tkn    = ws;                    // 2048*768
    float* lab    = tkn    + 2048*768;     // 160*768
    float* scores = lab    + 160*768;      // 2048*160
    float* attn   = scores + 2048*160;     // 2048*160
    float* b_in   = attn   + 2048*160;     // 40960
    float* q2c    = b_in   + 40960;        // 160*768

    dim3 blk(256);
    // tkn = token @ W1^T              [2048,768] x [768,768]
    gemm_nt_f32_wmma<<<dim3(16, 12), blk, 0, stream>>>(token, W1, tkn, 2048, 768, 768);
    // lab = label @ W2^T              [160,300]  x [300,768]
    gemm_nt_f32_wmma<<<dim3(2, 12),  blk, 0, stream>>>(label, W2, lab, 160, 768, 300);
    // scores = tkn @ lab^T            [2048,768] x [768,160]
    gemm_nt_f32_wmma<<<dim3(16, 3),  blk, 0, stream>>>(tkn, lab, scores, 2048, 160, 768);
    // masked softmax over LS + b_in
    attn_softmax_kernel<<<dim3(160), blk, 0, stream>>>(scores, lmask, imask, attn, b_in);
    // softmax over S + q2c
    q2c_kernel<<<dim3(160), blk, 0, stream>>>(b_in, tkn, q2c);
    // out = tanh(fused @ W5^T + b5)   [40960,3072] x [3072,768]  (dominant)
    fused_w5_tanh_wmma<<<dim3(320, 12), blk, 0, stream>>>(tkn, lab, attn, q2c, W5, b5, out);
}